// GridNeRF_55465207661097
// MI455X (gfx1250) — compile-verified
//
#include <hip/hip_runtime.h>
#include <stdint.h>

#define NRAYS 65536
#define NCOARSE 64
#define NFINE 128
#define NSAMP 192
#define RPB 8  // rays per block, one wave32 per ray

typedef unsigned int v4u __attribute__((ext_vector_type(4)));
typedef int v8i __attribute__((ext_vector_type(8)));
typedef int v4i __attribute__((ext_vector_type(4)));

// LDS byte offset of a __shared__ object: low 32 bits of the flat LDS-aperture address.
__device__ __forceinline__ unsigned lds_off(const void* p) {
  return (unsigned)(uintptr_t)p;
}

// Issue a TDM DMA: n contiguous f32 from global -> LDS (1-D tile descriptor, ISA sec 8).
__device__ __forceinline__ void tdm_load_row_f32(const float* gsrc, unsigned ldsByte, unsigned n) {
  unsigned long long ga = (unsigned long long)(uintptr_t)gsrc;
  v4u g0;
  g0[0] = 1u;                                              // count=1 valid descriptor
  g0[1] = ldsByte;                                         // lds_addr
  g0[2] = (unsigned)(ga & 0xFFFFFFFFull);                  // global_addr lo
  g0[3] = (unsigned)((ga >> 32) & 0x1FFFFFFull) | (2u << 30); // global_addr hi | type=2
  unsigned td0 = n, td1 = 1u, tile0 = n;                   // tensor dim0=n, 1 row; tile dim0=n
  unsigned long long s0 = n;                               // dim0 stride (unused for 1D tile)
  v8i g1;
  g1[0] = (int)(2u << 16);                                 // data_size=2 (4 bytes), no barrier/iterate/pad
  g1[1] = (int)((td0 & 0xFFFFu) << 16);                    // tensor_dim0[15:0] at bit48
  g1[2] = (int)(((td0 >> 16) & 0xFFFFu) | ((td1 & 0xFFFFu) << 16));
  g1[3] = (int)(((td1 >> 16) & 0xFFFFu) | ((tile0 & 0xFFFFu) << 16)); // tile_dim0 at bit112
  g1[4] = 0;                                               // tile_dim1=0 (1D), tile_dim2=0
  g1[5] = (int)(s0 & 0xFFFFFFFFull);                       // tensor_dim0_stride lo
  g1[6] = (int)((s0 >> 32) & 0xFFFFu);                     // stride hi | dim1_stride lo = 0
  g1[7] = 0;
  v4i z4 = {0, 0, 0, 0};
  v8i z8 = {0, 0, 0, 0, 0, 0, 0, 0};
  __builtin_amdgcn_tensor_load_to_lds(g0, g1, z4, z4, z8, 0);
}

__device__ __forceinline__ float wscan_add(float v, int lane) {
#pragma unroll
  for (int off = 1; off < 32; off <<= 1) {
    float t = __shfl_up(v, off, 32);
    if (lane >= off) v += t;
  }
  return v;
}
__device__ __forceinline__ unsigned wscan_addu(unsigned v, int lane) {
#pragma unroll
  for (int off = 1; off < 32; off <<= 1) {
    unsigned t = __shfl_up(v, off, 32);
    if (lane >= off) v += t;
  }
  return v;
}
__device__ __forceinline__ float wscan_mul(float v, int lane) {
#pragma unroll
  for (int off = 1; off < 32; off <<= 1) {
    float t = __shfl_up(v, off, 32);
    if (lane >= off) v *= t;
  }
  return v;
}
__device__ __forceinline__ float wreduce_add(float v) {
#pragma unroll
  for (int off = 16; off; off >>= 1) v += __shfl_xor(v, off, 32);
  return v;
}

__global__ __launch_bounds__(256) void nerf_hier_render(
    const float* __restrict__ RO, const float* __restrict__ RD,
    const float* __restrict__ CW, const float* __restrict__ CT,
    const float* __restrict__ U, const float* __restrict__ COL,
    const float* __restrict__ DEN, float* __restrict__ OUT) {
  __shared__ float sCol[RPB][576];          // TDM target: colors row
  __shared__ float sDen[RPB][NSAMP];        // TDM target: densities row
  __shared__ float sCdf[RPB][NCOARSE + 2];  // 65 cdf entries (+pad)
  __shared__ float sCt[RPB][NCOARSE];
  __shared__ unsigned sHist[RPB][NCOARSE];
  __shared__ unsigned sBase[RPB][NCOARSE];
  __shared__ float sComb[RPB][NSAMP];

  const int wid = __builtin_amdgcn_readfirstlane((int)(threadIdx.x >> 5));
  const int lane = (int)(threadIdx.x & 31);
  const int ray = (int)blockIdx.x * RPB + wid;

  // Kick off async TDM copies of this ray's colors/densities; consumed in phase 4.
  tdm_load_row_f32(COL + (size_t)ray * 576, lds_off(&sCol[wid][0]), 576);
  tdm_load_row_f32(DEN + (size_t)ray * NSAMP, lds_off(&sDen[wid][0]), NSAMP);

  // ---------- Phase 1: CDF over coarse weights (wave32 scan) ----------
  float2 wv = *(const float2*)(CW + (size_t)ray * NCOARSE + 2 * lane);
  float2 tv = *(const float2*)(CT + (size_t)ray * NCOARSE + 2 * lane);
  float w0 = wv.x + 1e-5f, w1 = wv.y + 1e-5f;
  float pair = w0 + w1;
  float incl = wscan_add(pair, lane);
  float total = __shfl(incl, 31, 32);
  float invT = 1.0f / total;
  float excl = incl - pair;
  if (lane == 0) sCdf[wid][0] = 0.0f;
  sCdf[wid][2 * lane + 1] = (excl + w0) * invT;
  sCdf[wid][2 * lane + 2] = incl * invT;
  sCt[wid][2 * lane] = tv.x;
  sCt[wid][2 * lane + 1] = tv.y;
  sHist[wid][2 * lane] = 0u;
  sHist[wid][2 * lane + 1] = 0u;
  __syncthreads();

  // ---------- Phase 2: inverse-CDF fine sampling (4 samples/lane) ----------
  const float* cdf = &sCdf[wid][0];
  const float* ct = &sCt[wid][0];
  float4 u4 = *(const float4*)(U + (size_t)ray * NFINE + 4 * lane);
  float uf[4] = {u4.x, u4.y, u4.z, u4.w};
  int bj[4];
  float fv[4];
  unsigned slot[4];
#pragma unroll
  for (int j = 0; j < 4; ++j) {
    float u = uf[j];
    int lo = 0, hi = NCOARSE + 1;  // searchsorted(right) over 65 cdf entries
    while (lo < hi) {
      int mid = (lo + hi) >> 1;
      if (cdf[mid] <= u) lo = mid + 1; else hi = mid;
    }
    int below = lo - 1;
    below = below < 0 ? 0 : (below > NCOARSE - 1 ? NCOARSE - 1 : below);
    int above = lo > NCOARSE - 1 ? NCOARSE - 1 : lo;
    float cb = cdf[below], ca = cdf[above];
    float tb = ct[below], ta = ct[above];
    float den = ca - cb;
    if (den < 1e-5f) den = 1.0f;
    float tt = (u - cb) / den;
    fv[j] = tb + tt * (ta - tb);
    bj[j] = below;  // fine value lies in [ct[below], ct[below+1])
    slot[j] = atomicAdd(&sHist[wid][below], 1u);  // arrival slot within bin
  }
  __syncthreads();

  // ---------- Phase 3: O(S) merge into sorted order via bin histogram ----------
  unsigned h0 = sHist[wid][2 * lane], h1 = sHist[wid][2 * lane + 1];
  unsigned hpair = h0 + h1;
  unsigned hincl = wscan_addu(hpair, lane);
  unsigned hexcl = hincl - hpair;  // #fines in bins < 2*lane
  unsigned H0 = hexcl, H1 = hexcl + h0;
  sBase[wid][2 * lane] = (unsigned)(2 * lane) + 1u + H0;      // fine run start for bin b
  sBase[wid][2 * lane + 1] = (unsigned)(2 * lane + 1) + 1u + H1;
  sComb[wid][2 * lane + H0] = tv.x;       // coarse rank = b + H(b)
  sComb[wid][2 * lane + 1 + H1] = tv.y;
  __syncthreads();
#pragma unroll
  for (int j = 0; j < 4; ++j)
    sComb[wid][sBase[wid][bj[j]] + slot[j]] = fv[j];
  __syncthreads();
  // sort each (tiny, avg ~2-element) same-bin run; 2 bins per lane
#pragma unroll
  for (int bb = 0; bb < 2; ++bb) {
    int st = (int)sBase[wid][2 * lane + bb];
    int ln = (int)(bb ? h1 : h0);
    float* a = &sComb[wid][st];
    for (int i = 1; i < ln; ++i) {
      float v = a[i];
      int j2 = i - 1;
      while (j2 >= 0 && a[j2] > v) { a[j2 + 1] = a[j2]; --j2; }
      a[j2 + 1] = v;
    }
  }
  __syncthreads();

  // ---------- Phase 4: volume rendering (needs TDM data now) ----------
  __builtin_amdgcn_s_wait_tensorcnt(0);
  asm volatile("" ::: "memory");  // TDM wrote LDS behind the compiler's back

  float ox = RO[(size_t)ray * 3 + 0], oy = RO[(size_t)ray * 3 + 1], oz = RO[(size_t)ray * 3 + 2];
  float dx = RD[(size_t)ray * 3 + 0], dy = RD[(size_t)ray * 3 + 1], dz = RD[(size_t)ray * 3 + 2];
  float dnorm = sqrtf(dx * dx + dy * dy + dz * dz);

  float t6[6], alpha6[6], eloc[6];
  float Plocal = 1.0f;
#pragma unroll
  for (int k = 0; k < 6; ++k) {
    int s = 6 * lane + k;
    float ts = sComb[wid][s];
    float dist = ((s == NSAMP - 1) ? 1e10f : (sComb[wid][s + 1] - ts)) * dnorm;
    float ex = __expf(-sDen[wid][s] * dist);
    t6[k] = ts;
    alpha6[k] = 1.0f - ex;
    float q = ex + 1e-10f;
    eloc[k] = Plocal;  // local exclusive prefix product
    Plocal *= q;
  }
  float inclP = wscan_mul(Plocal, lane);   // wave-level exclusive cumprod
  float Texcl = __shfl_up(inclP, 1, 32);
  if (lane == 0) Texcl = 1.0f;

  float accR = 0.f, accG = 0.f, accB = 0.f, accD = 0.f, accO = 0.f;
  float fp[18];
#pragma unroll
  for (int k = 0; k < 6; ++k) {
    int s = 6 * lane + k;
    float w = alpha6[k] * (Texcl * eloc[k]);
    accR += w * sCol[wid][3 * s + 0];
    accG += w * sCol[wid][3 * s + 1];
    accB += w * sCol[wid][3 * s + 2];
    accD += w * t6[k];
    accO += w;
    fp[3 * k + 0] = ox + dx * t6[k];
    fp[3 * k + 1] = oy + dy * t6[k];
    fp[3 * k + 2] = oz + dz * t6[k];
  }
  accR = wreduce_add(accR);
  accG = wreduce_add(accG);
  accB = wreduce_add(accB);
  accD = wreduce_add(accD);
  accO = wreduce_add(accO);

  float* outDep = OUT + (size_t)3 * NRAYS;
  float* outOpa = OUT + (size_t)4 * NRAYS;
  float* outFP = OUT + (size_t)5 * NRAYS;
  if (lane == 0) {
    OUT[(size_t)ray * 3 + 0] = accR;
    OUT[(size_t)ray * 3 + 1] = accG;
    OUT[(size_t)ray * 3 + 2] = accB;
    outDep[ray] = accD;
    outOpa[ray] = accO;
  }
  // each lane owns 18 contiguous floats of fine_points -> 9 coalesced b64 stores
  float2* fpb = (float2*)(outFP + (size_t)ray * 576 + lane * 18);
#pragma unroll
  for (int m = 0; m < 9; ++m) fpb[m] = make_float2(fp[2 * m], fp[2 * m + 1]);
}

extern "C" void kernel_launch(void* const* d_in, const int* in_sizes, int n_in,
                              void* d_out, int out_size, void* d_ws, size_t ws_size,
                              hipStream_t stream) {
  (void)in_sizes; (void)n_in; (void)d_ws; (void)ws_size; (void)out_size;
  const float* RO = (const float*)d_in[0];
  const float* RD = (const float*)d_in[1];
  const float* CW = (const float*)d_in[2];
  const float* CT = (const float*)d_in[3];
  const float* U = (const float*)d_in[4];
  const float* COL = (const float*)d_in[5];
  const float* DEN = (const float*)d_in[6];
  float* OUT = (float*)d_out;
  dim3 grid(NRAYS / RPB), block(256);
  hipLaunchKernelGGL(nerf_hier_render, grid, block, 0, stream,
                     RO, RD, CW, CT, U, COL, DEN, OUT);
}